// WaveNet_14783277433303
// MI455X (gfx1250) — compile-verified
//
#include <hip/hip_runtime.h>
#include <math.h>

// ---------------- problem constants (match reference) ----------------
#define Hh   128      // hidden channels
#define Cc   256      // classes / input channels
#define NBk  8        // blocks
#define Bb   16       // batch
#define Tt   8192     // time
#define NT   64       // tokens per tile
#define LSTR 65       // plain LDS stride (output staging)
#define PSTR 160      // paired LDS stride (dwords per channel-pair row);
                      // 160 % 64 == 32 -> wave32 b64 loads hit all 64 banks

typedef float v2f __attribute__((ext_vector_type(2)));
typedef float v4f __attribute__((ext_vector_type(4)));
typedef float v8f __attribute__((ext_vector_type(8)));

// D = A(16x4 f32) * B(4x16 f32) + C  on the gfx1250 matrix pipe
#define WMMA4(a, b, c) \
  __builtin_amdgcn_wmma_f32_16x16x4_f32(false, (a), false, (b), (short)0, (c), false, false)

// paired layout: element (channel c, column col) of a staged tile
__device__ __forceinline__ int pidx(int c, int col) {
  return (c >> 1) * PSTR + (col << 1) + (c & 1);
}

__device__ __forceinline__ float gate_fn(float v) {
  return tanhf(v) * (1.0f / (1.0f + __expf(-v)));  // tanh * sigmoid
}

// =====================================================================
// Weight repack: K=2 conv weights [row][c][tap] -> tap-major [tap][row][c]
// so A fragments are contiguous 8B loads.
// =====================================================================
__global__ __launch_bounds__(256) void k_repack_in(
    const float* __restrict__ w_in, float* __restrict__ wpk) {
  int i = blockIdx.x * 256 + threadIdx.x;  // over H*C
  if (i < Hh * Cc) {
    wpk[i] = w_in[(size_t)i * 2 + 0];
    wpk[Hh * Cc + i] = w_in[(size_t)i * 2 + 1];
  }
}
__global__ __launch_bounds__(256) void k_repack_dil(
    const float* __restrict__ w_dil, float* __restrict__ wpk) {
  int i = blockIdx.x * 256 + threadIdx.x;  // over NB*H*H
  if (i < NBk * Hh * Hh) {
    int blk = i / (Hh * Hh), rc = i % (Hh * Hh);
    float* dst = wpk + (size_t)blk * 2 * Hh * Hh;
    dst[rc] = w_dil[(size_t)i * 2 + 0];
    dst[Hh * Hh + rc] = w_dil[(size_t)i * 2 + 1];
  }
}

// =====================================================================
// Kernel 1: input causal conv  x[B,C,T] -> h[B,H,T], K=2, dilation=1
// =====================================================================
__global__ __launch_bounds__(256) void k_conv_in(
    const float* __restrict__ x, const float* __restrict__ wpk,
    const float* __restrict__ b_in, float* __restrict__ h_out) {
  __shared__ float xs[(Cc / 2) * PSTR];  // paired; col j = token t0-1+j, j=0..64
  __shared__ float ob[Hh * LSTR];

  const int tid  = threadIdx.x;
  const int tile = blockIdx.x;
  const int b    = tile / (Tt / NT);
  const int t0   = (tile % (Tt / NT)) * NT;
  const float* xb = x + (size_t)b * Cc * Tt;

  for (int idx = tid; idx < Cc * 16; idx += 256) {
    int row = idx >> 4, c4 = (idx & 15) << 2;
    v4f v = *(const v4f*)(xb + (size_t)row * Tt + t0 + c4);
    xs[pidx(row, c4 + 1)] = v.x;
    xs[pidx(row, c4 + 2)] = v.y;
    xs[pidx(row, c4 + 3)] = v.z;
    xs[pidx(row, c4 + 4)] = v.w;
  }
  for (int row = tid; row < Cc; row += 256)
    xs[pidx(row, 0)] = (t0 > 0) ? xb[(size_t)row * Tt + t0 - 1] : 0.0f;
  __syncthreads();

  const int wave = tid >> 5, lane = tid & 31;
  const int half = lane >> 4, l16 = lane & 15;
  const int mq = wave & 3;        // M group: rows [32*mq, 32*mq+31]
  const int nh = wave >> 2;       // N half: cols [32*nh, 32*nh+31]
  const int colbase = nh * 32;
  const int rowA0 = mq * 32 + l16, rowA1 = rowA0 + 16;

  v8f acc[2][2];
#pragma unroll
  for (int mi = 0; mi < 2; mi++)
#pragma unroll
    for (int n = 0; n < 2; n++)
#pragma unroll
      for (int j = 0; j < 8; j++)
        acc[mi][n][j] = b_in[mq * 32 + mi * 16 + j + half * 8];

  const float* wp0 = wpk;              // tap0 (token t-1)
  const float* wp1 = wpk + Hh * Cc;    // tap1 (token t)
  for (int k0 = 0; k0 < Cc; k0 += 4) {
    int ce = k0 + half * 2;
    v2f a00 = *(const v2f*)(wp0 + (size_t)rowA0 * Cc + ce);
    v2f a01 = *(const v2f*)(wp1 + (size_t)rowA0 * Cc + ce);
    v2f a10 = *(const v2f*)(wp0 + (size_t)rowA1 * Cc + ce);
    v2f a11 = *(const v2f*)(wp1 + (size_t)rowA1 * Cc + ce);
    const float* xr = &xs[(ce >> 1) * PSTR];
#pragma unroll
    for (int n = 0; n < 2; n++) {
      int col = colbase + n * 16 + l16;            // token index
      v2f bp = *(const v2f*)(xr + col * 2);        // xs col = token (t-1 tap)
      v2f bc = *(const v2f*)(xr + (col + 1) * 2);  // xs col = token+1 (t tap)
      acc[0][n] = WMMA4(a00, bp, acc[0][n]);
      acc[0][n] = WMMA4(a01, bc, acc[0][n]);
      acc[1][n] = WMMA4(a10, bp, acc[1][n]);
      acc[1][n] = WMMA4(a11, bc, acc[1][n]);
    }
  }

#pragma unroll
  for (int mi = 0; mi < 2; mi++)
#pragma unroll
    for (int n = 0; n < 2; n++)
#pragma unroll
      for (int j = 0; j < 8; j++) {
        int row = mq * 32 + mi * 16 + j + half * 8;
        ob[row * LSTR + colbase + n * 16 + l16] = acc[mi][n][j];
      }
  __syncthreads();

  float* hb = h_out + (size_t)b * Hh * Tt + t0;
  for (int idx = tid; idx < Hh * 16; idx += 256) {
    int row = idx >> 4, c4 = (idx & 15) << 2;
    float* p = &ob[row * LSTR + c4];
    v4f v = {p[0], p[1], p[2], p[3]};
    *(v4f*)(hb + (size_t)row * Tt + c4) = v;
  }
}

// =====================================================================
// Kernel 2: one WaveNet block (dilated conv + gate + 1x1 + skip/res)
// =====================================================================
__global__ __launch_bounds__(256) void k_block(
    const float* __restrict__ h_in, float* __restrict__ h_out,
    float* __restrict__ skip, const float* __restrict__ wpk_dil,
    const float* __restrict__ b_dil, const float* __restrict__ w_res,
    const float* __restrict__ b_res, int dil, int first) {
  __shared__ float cur[(Hh / 2) * PSTR];  // h tile [t0, t0+63]       (paired)
  __shared__ float prv[(Hh / 2) * PSTR];  // h tile [t0-d, ...]; later res
  __shared__ float gat[(Hh / 2) * PSTR];  // gated activations        (paired)

  const int tid  = threadIdx.x;
  const int tile = blockIdx.x;
  const int b    = tile / (Tt / NT);
  const int t0   = (tile % (Tt / NT)) * NT;
  const float* hbin = h_in + (size_t)b * Hh * Tt;

  for (int idx = tid; idx < Hh * 16; idx += 256) {
    int row = idx >> 4, c4 = (idx & 15) << 2;
    v4f v = *(const v4f*)(hbin + (size_t)row * Tt + t0 + c4);
    cur[pidx(row, c4 + 0)] = v.x;
    cur[pidx(row, c4 + 1)] = v.y;
    cur[pidx(row, c4 + 2)] = v.z;
    cur[pidx(row, c4 + 3)] = v.w;
  }
  for (int idx = tid; idx < Hh * NT; idx += 256) {
    int row = idx >> 6, cc = idx & 63;
    int tt = t0 + cc - dil;
    prv[pidx(row, cc)] = (tt >= 0) ? hbin[(size_t)row * Tt + tt] : 0.0f;
  }
  __syncthreads();

  const int wave = tid >> 5, lane = tid & 31;
  const int half = lane >> 4, l16 = lane & 15;
  const int mq = wave & 3, nh = wave >> 2;
  const int colbase = nh * 32;
  const int rowA0 = mq * 32 + l16, rowA1 = rowA0 + 16;

  // ---- GEMM1: dilated conv ----
  v8f acc[2][2];
#pragma unroll
  for (int mi = 0; mi < 2; mi++)
#pragma unroll
    for (int n = 0; n < 2; n++)
#pragma unroll
      for (int j = 0; j < 8; j++)
        acc[mi][n][j] = b_dil[mq * 32 + mi * 16 + j + half * 8];

  const float* wd0 = wpk_dil;               // tap0 -> t-d
  const float* wd1 = wpk_dil + Hh * Hh;     // tap1 -> t
  for (int k0 = 0; k0 < Hh; k0 += 4) {
    int ce = k0 + half * 2;
    v2f a00 = *(const v2f*)(wd0 + (size_t)rowA0 * Hh + ce);
    v2f a01 = *(const v2f*)(wd1 + (size_t)rowA0 * Hh + ce);
    v2f a10 = *(const v2f*)(wd0 + (size_t)rowA1 * Hh + ce);
    v2f a11 = *(const v2f*)(wd1 + (size_t)rowA1 * Hh + ce);
    const float* pr = &prv[(ce >> 1) * PSTR];
    const float* cr = &cur[(ce >> 1) * PSTR];
#pragma unroll
    for (int n = 0; n < 2; n++) {
      int col = colbase + n * 16 + l16;
      v2f bp = *(const v2f*)(pr + col * 2);
      v2f bc = *(const v2f*)(cr + col * 2);
      acc[0][n] = WMMA4(a00, bp, acc[0][n]);
      acc[0][n] = WMMA4(a01, bc, acc[0][n]);
      acc[1][n] = WMMA4(a10, bp, acc[1][n]);
      acc[1][n] = WMMA4(a11, bc, acc[1][n]);
    }
  }

  // ---- gate -> LDS (paired) ----
#pragma unroll
  for (int mi = 0; mi < 2; mi++)
#pragma unroll
    for (int n = 0; n < 2; n++)
#pragma unroll
      for (int j = 0; j < 8; j++) {
        int row = mq * 32 + mi * 16 + j + half * 8;
        gat[pidx(row, colbase + n * 16 + l16)] = gate_fn(acc[mi][n][j]);
      }
  __syncthreads();

  // ---- GEMM2: 1x1 residual conv ----
  v8f acc2[2][2];
#pragma unroll
  for (int mi = 0; mi < 2; mi++)
#pragma unroll
    for (int n = 0; n < 2; n++)
#pragma unroll
      for (int j = 0; j < 8; j++)
        acc2[mi][n][j] = b_res[mq * 32 + mi * 16 + j + half * 8];

  for (int k0 = 0; k0 < Hh; k0 += 4) {
    int ce = k0 + half * 2;
    v2f a0 = *(const v2f*)(w_res + (size_t)rowA0 * Hh + ce);
    v2f a1 = *(const v2f*)(w_res + (size_t)rowA1 * Hh + ce);
    const float* gr = &gat[(ce >> 1) * PSTR];
#pragma unroll
    for (int n = 0; n < 2; n++) {
      int col = colbase + n * 16 + l16;
      v2f bb = *(const v2f*)(gr + col * 2);
      acc2[0][n] = WMMA4(a0, bb, acc2[0][n]);
      acc2[1][n] = WMMA4(a1, bb, acc2[1][n]);
    }
  }

  // stash res into prv (dead after GEMM1; all waves past barrier above)
#pragma unroll
  for (int mi = 0; mi < 2; mi++)
#pragma unroll
    for (int n = 0; n < 2; n++)
#pragma unroll
      for (int j = 0; j < 8; j++) {
        int row = mq * 32 + mi * 16 + j + half * 8;
        prv[pidx(row, colbase + n * 16 + l16)] = acc2[mi][n][j];
      }
  __syncthreads();

  // ---- epilogue: h_out = cur + res ; skip (+)= res ----
  float* hbo = h_out + (size_t)b * Hh * Tt + t0;
  float* sbo = skip + (size_t)b * Hh * Tt + t0;
  for (int idx = tid; idx < Hh * 16; idx += 256) {
    int row = idx >> 4, c4 = (idx & 15) << 2;
    v4f r, h;
#pragma unroll
    for (int i = 0; i < 4; i++) {
      float rv = prv[pidx(row, c4 + i)];
      r[i] = rv;
      h[i] = cur[pidx(row, c4 + i)] + rv;
    }
    *(v4f*)(hbo + (size_t)row * Tt + c4) = h;
    v4f s;
    if (first) {
      s = r;
    } else {
      v4f so = *(const v4f*)(sbo + (size_t)row * Tt + c4);
      s.x = so.x + r.x; s.y = so.y + r.y; s.z = so.z + r.z; s.w = so.w + r.w;
    }
    *(v4f*)(sbo + (size_t)row * Tt + c4) = s;
  }
}

// =====================================================================
// Kernel 3: relu(skip) -> 1x1 (H->H) -> relu -> 1x1 (H->C) => logits
// =====================================================================
__global__ __launch_bounds__(256) void k_post(
    const float* __restrict__ skip, const float* __restrict__ w_p1,
    const float* __restrict__ b_p1, const float* __restrict__ w_p2,
    const float* __restrict__ b_p2, float* __restrict__ out) {
  __shared__ float sb[(Hh / 2) * PSTR];  // relu(skip), paired
  __shared__ float zb[(Hh / 2) * PSTR];  // relu(z), paired

  const int tid  = threadIdx.x;
  const int tile = blockIdx.x;
  const int b    = tile / (Tt / NT);
  const int t0   = (tile % (Tt / NT)) * NT;
  const float* sk = skip + (size_t)b * Hh * Tt;

  for (int idx = tid; idx < Hh * 16; idx += 256) {
    int row = idx >> 4, c4 = (idx & 15) << 2;
    v4f v = *(const v4f*)(sk + (size_t)row * Tt + t0 + c4);
    sb[pidx(row, c4 + 0)] = fmaxf(v.x, 0.f);
    sb[pidx(row, c4 + 1)] = fmaxf(v.y, 0.f);
    sb[pidx(row, c4 + 2)] = fmaxf(v.z, 0.f);
    sb[pidx(row, c4 + 3)] = fmaxf(v.w, 0.f);
  }
  __syncthreads();

  const int wave = tid >> 5, lane = tid & 31;
  const int half = lane >> 4, l16 = lane & 15;
  const int mq = wave & 3, nh = wave >> 2;
  const int colbase = nh * 32;

  // ---- GEMM1: z = relu(W_p1 @ relu(skip) + b1), M=128 ----
  {
    const int rowA0 = mq * 32 + l16, rowA1 = rowA0 + 16;
    v8f acc[2][2];
#pragma unroll
    for (int mi = 0; mi < 2; mi++)
#pragma unroll
      for (int n = 0; n < 2; n++)
#pragma unroll
        for (int j = 0; j < 8; j++)
          acc[mi][n][j] = b_p1[mq * 32 + mi * 16 + j + half * 8];

    for (int k0 = 0; k0 < Hh; k0 += 4) {
      int ce = k0 + half * 2;
      v2f a0 = *(const v2f*)(w_p1 + (size_t)rowA0 * Hh + ce);
      v2f a1 = *(const v2f*)(w_p1 + (size_t)rowA1 * Hh + ce);
      const float* sr = &sb[(ce >> 1) * PSTR];
#pragma unroll
      for (int n = 0; n < 2; n++) {
        int col = colbase + n * 16 + l16;
        v2f bb = *(const v2f*)(sr + col * 2);
        acc[0][n] = WMMA4(a0, bb, acc[0][n]);
        acc[1][n] = WMMA4(a1, bb, acc[1][n]);
      }
    }
#pragma unroll
    for (int mi = 0; mi < 2; mi++)
#pragma unroll
      for (int n = 0; n < 2; n++)
#pragma unroll
        for (int j = 0; j < 8; j++) {
          int row = mq * 32 + mi * 16 + j + half * 8;
          zb[pidx(row, colbase + n * 16 + l16)] = fmaxf(acc[mi][n][j], 0.f);
        }
  }
  __syncthreads();

  // ---- GEMM2: logits = W_p2 @ z + b2, M=256 (4 m-tiles/wave, 2 passes) ----
  float* ob = out + (size_t)b * Cc * Tt + t0;
  for (int pass = 0; pass < 2; pass++) {
    const int rbase = mq * 64 + pass * 32;
    const int rowA0 = rbase + l16, rowA1 = rowA0 + 16;
    v8f acc[2][2];
#pragma unroll
    for (int mi = 0; mi < 2; mi++)
#pragma unroll
      for (int n = 0; n < 2; n++)
#pragma unroll
        for (int j = 0; j < 8; j++)
          acc[mi][n][j] = b_p2[rbase + mi * 16 + j + half * 8];

    for (int k0 = 0; k0 < Hh; k0 += 4) {
      int ce = k0 + half * 2;
      v2f a0 = *(const v2f*)(w_p2 + (size_t)rowA0 * Hh + ce);
      v2f a1 = *(const v2f*)(w_p2 + (size_t)rowA1 * Hh + ce);
      const float* zr = &zb[(ce >> 1) * PSTR];
#pragma unroll
      for (int n = 0; n < 2; n++) {
        int col = colbase + n * 16 + l16;
        v2f bb = *(const v2f*)(zr + col * 2);
        acc[0][n] = WMMA4(a0, bb, acc[0][n]);
        acc[1][n] = WMMA4(a1, bb, acc[1][n]);
      }
    }
#pragma unroll
    for (int mi = 0; mi < 2; mi++)
#pragma unroll
      for (int n = 0; n < 2; n++)
#pragma unroll
        for (int j = 0; j < 8; j++) {
          int row = rbase + mi * 16 + j + half * 8;
          ob[(size_t)row * Tt + colbase + n * 16 + l16] = acc[mi][n][j];
        }
  }
}

// =====================================================================
// Kernel 4: in-place softmax over axis 0 (batch) per (c,t)
// =====================================================================
__global__ __launch_bounds__(256) void k_softmax(float* __restrict__ out) {
  const size_t ct = (size_t)blockIdx.x * 256 + threadIdx.x;
  const size_t stride = (size_t)Cc * Tt;
  float v[Bb];
  float m = -INFINITY;
#pragma unroll
  for (int b = 0; b < Bb; b++) {
    v[b] = out[(size_t)b * stride + ct];
    m = fmaxf(m, v[b]);
  }
  float s = 0.f;
#pragma unroll
  for (int b = 0; b < Bb; b++) {
    v[b] = __expf(v[b] - m);
    s += v[b];
  }
  float inv = 1.f / s;
#pragma unroll
  for (int b = 0; b < Bb; b++) out[(size_t)b * stride + ct] = v[b] * inv;
}

// =====================================================================
extern "C" void kernel_launch(void* const* d_in, const int* in_sizes, int n_in,
                              void* d_out, int out_size, void* d_ws,
                              size_t ws_size, hipStream_t stream) {
  const float* x     = (const float*)d_in[0];
  const float* w_in  = (const float*)d_in[1];
  const float* b_in  = (const float*)d_in[2];
  const float* w_dil = (const float*)d_in[3];
  const float* b_dil = (const float*)d_in[4];
  const float* w_res = (const float*)d_in[5];
  const float* b_res = (const float*)d_in[6];
  const float* w_p1  = (const float*)d_in[7];
  const float* b_p1  = (const float*)d_in[8];
  const float* w_p2  = (const float*)d_in[9];
  const float* b_p2  = (const float*)d_in[10];
  float* out = (float*)d_out;

  // workspace carve: hA | hB | skip | wpk_in | wpk_dil
  const size_t hsz = (size_t)Bb * Hh * Tt;  // 16.78M floats each
  float* hA = (float*)d_ws;
  float* hB = hA + hsz;
  float* sk = hB + hsz;
  float* wpk_in = sk + hsz;                       // 2*H*C floats
  float* wpk_dil = wpk_in + (size_t)2 * Hh * Cc;  // NB*2*H*H floats

  const int ntiles = Bb * (Tt / NT);  // 2048
  dim3 grid(ntiles), blk(256);

  k_repack_in<<<(Hh * Cc + 255) / 256, blk, 0, stream>>>(w_in, wpk_in);
  k_repack_dil<<<(NBk * Hh * Hh + 255) / 256, blk, 0, stream>>>(w_dil, wpk_dil);

  k_conv_in<<<grid, blk, 0, stream>>>(x, wpk_in, b_in, hA);

  float* hin = hA;
  float* hout = hB;
  for (int i = 0; i < NBk; i++) {
    k_block<<<grid, blk, 0, stream>>>(
        hin, hout, sk, wpk_dil + (size_t)i * 2 * Hh * Hh,
        b_dil + (size_t)i * Hh, w_res + (size_t)i * Hh * Hh,
        b_res + (size_t)i * Hh, 1 << i, (i == 0) ? 1 : 0);
    float* t = hin; hin = hout; hout = t;
  }

  k_post<<<grid, blk, 0, stream>>>(sk, w_p1, b_p1, w_p2, b_p2, out);
  k_softmax<<<((size_t)Cc * Tt) / 256, blk, 0, stream>>>(out);
}